// ClassifierHetero_28956669509884
// MI455X (gfx1250) — compile-verified
//
#include <hip/hip_runtime.h>

typedef float v2f __attribute__((ext_vector_type(2)));
typedef float v8f __attribute__((ext_vector_type(8)));

#define HID  128
#define NCLS 10
#define BGR  64

// Workspace layout (floats):
//   [0..63]    comp_sum     [64..127]  comp_cnt
//   [128..191] port_sum0    [192..255] port_sum1   [256..319] port_cnt
//   [320..383] net_sum      [384..447] net_cnt
#define WS_FLOATS 448

__global__ void zero_ws_kernel(float* __restrict__ ws) {
    int i = blockIdx.x * blockDim.x + threadIdx.x;
    if (i < WS_FLOATS) ws[i] = 0.0f;
}

// Segment sums + counts for the three node types, grouped by graph id.
// gids are sorted, so lanes in a wave mostly hit the same LDS bucket;
// per-block LDS accumulation, then 448 global atomics per block.
__global__ __launch_bounds__(256) void seg_accum_kernel(
    const float* __restrict__ h_comp, const float* __restrict__ h_port,
    const float* __restrict__ h_net,
    const int* __restrict__ gid_comp, const int* __restrict__ gid_port,
    const int* __restrict__ gid_net,
    int nc, int np, int nn, float* __restrict__ ws) {
    __shared__ float s[WS_FLOATS];
    for (int i = threadIdx.x; i < WS_FLOATS; i += blockDim.x) s[i] = 0.0f;
    __syncthreads();

    const int stride = gridDim.x * blockDim.x;
    const int t0 = blockIdx.x * blockDim.x + threadIdx.x;

    for (int i = t0; i < nc; i += stride) {
        int g = gid_comp[i];
        atomicAdd(&s[g], h_comp[i]);
        atomicAdd(&s[64 + g], 1.0f);
    }
    for (int i = t0; i < np; i += stride) {
        int g = gid_port[i];
        atomicAdd(&s[128 + g], h_port[2 * i + 0]);
        atomicAdd(&s[192 + g], h_port[2 * i + 1]);
        atomicAdd(&s[256 + g], 1.0f);
    }
    for (int i = t0; i < nn; i += stride) {
        int g = gid_net[i];
        atomicAdd(&s[320 + g], h_net[i]);
        atomicAdd(&s[384 + g], 1.0f);
    }
    __syncthreads();
    for (int i = threadIdx.x; i < WS_FLOATS; i += blockDim.x)
        atomicAdd(&ws[i], s[i]);
}

static __device__ __forceinline__ v8f wmma_f32_k4(v2f a, v2f b, v8f c) {
    // V_WMMA_F32_16X16X4_F32: A 16x4 f32 (2 VGPRs), B 4x16 f32 (2 VGPRs),
    // C/D 16x16 f32 (8 VGPRs). 8-arg form: (neg_a, A, neg_b, B, c_mod, C, reuse_a, reuse_b)
    return __builtin_amdgcn_wmma_f32_16x16x4_f32(
        false, a, false, b, (short)0, c, false, false);
}

// Single workgroup (8 wave32): finalize graph means, then run the whole
// classifier MLP on the f32 WMMA pipe. Activations staged in LDS.
__global__ __launch_bounds__(256) void mlp_wmma_kernel(
    const float* __restrict__ ws,
    const float* __restrict__ Wc1, const float* __restrict__ bc1,
    const float* __restrict__ Wc2, const float* __restrict__ bc2,
    const float* __restrict__ Wc3, const float* __restrict__ bc3,
    float* __restrict__ out) {
    __shared__ float s_hg[BGR * 4];      // pooled features [64, 4]
    __shared__ float s_h[BGR * HID];     // activations     [64, 128]

    const int tid  = threadIdx.x;
    const int wave = tid >> 5;
    const int lane = tid & 31;
    const int lr   = lane & 15;          // row (A) / col (B,C/D) within a 16-tile
    const int lh   = lane >> 4;          // half-wave select

    // ---- hg = concat(mean_comp, mean_port0, mean_port1, mean_net) ----
    if (tid < BGR) {
        float cc = fmaxf(ws[64 + tid], 1.0f);
        float pc = fmaxf(ws[256 + tid], 1.0f);
        float nc = fmaxf(ws[384 + tid], 1.0f);
        s_hg[tid * 4 + 0] = ws[tid] / cc;
        s_hg[tid * 4 + 1] = ws[128 + tid] / pc;
        s_hg[tid * 4 + 2] = ws[192 + tid] / pc;
        s_hg[tid * 4 + 3] = ws[320 + tid] / nc;
    }
    __syncthreads();

    // ---- Layer 1: h1 = relu(hg[64,4] @ Wc1[4,128] + bc1), K=4 -> 1 WMMA/tile
    // 4 Mtiles x 8 Ntiles = 32 tiles; wave w owns tiles [4w, 4w+3].
    for (int j = 0; j < 4; ++j) {
        int t  = wave * 4 + j;
        int mt = t >> 3, nt = t & 7;
        v2f a, b;
        a.x = s_hg[(mt * 16 + lr) * 4 + lh * 2 + 0];
        a.y = s_hg[(mt * 16 + lr) * 4 + lh * 2 + 1];
        b.x = Wc1[(lh * 2 + 0) * HID + nt * 16 + lr];
        b.y = Wc1[(lh * 2 + 1) * HID + nt * 16 + lr];
        v8f acc = {};
        acc = wmma_f32_k4(a, b, acc);
        int col = nt * 16 + lr;
        float bias = bc1[col];
        for (int v = 0; v < 8; ++v) {
            int row = mt * 16 + v + 8 * lh;
            s_h[row * HID + col] = fmaxf(acc[v] + bias, 0.0f);
        }
    }
    __syncthreads();

    // ---- Layer 2: h2 = relu(h1[64,128] @ Wc2[128,128] + bc2), 32 K-steps/tile
    v8f r[4];
    for (int j = 0; j < 4; ++j) {
        int t  = wave * 4 + j;
        int mt = t >> 3, nt = t & 7;
        v8f acc = {};
        for (int kk = 0; kk < 32; ++kk) {
            int k0 = kk * 4 + lh * 2;
            v2f a, b;
            a.x = s_h[(mt * 16 + lr) * HID + k0 + 0];
            a.y = s_h[(mt * 16 + lr) * HID + k0 + 1];
            b.x = Wc2[(k0 + 0) * HID + nt * 16 + lr];
            b.y = Wc2[(k0 + 1) * HID + nt * 16 + lr];
            acc = wmma_f32_k4(a, b, acc);
        }
        float bias = bc2[nt * 16 + lr];
        for (int v = 0; v < 8; ++v) r[j][v] = fmaxf(acc[v] + bias, 0.0f);
    }
    __syncthreads();   // all waves finished reading h1
    for (int j = 0; j < 4; ++j) {
        int t  = wave * 4 + j;
        int mt = t >> 3, nt = t & 7;
        int col = nt * 16 + lr;
        for (int v = 0; v < 8; ++v)
            s_h[(mt * 16 + v + 8 * lh) * HID + col] = r[j][v];
    }
    __syncthreads();

    // ---- Layer 3: out = h2[64,128] @ Wc3[128,10] + bc3 (N padded 10->16)
    if (wave < 4) {                       // wave-uniform branch: EXEC stays full
        int mt = wave;
        v8f acc = {};
        for (int kk = 0; kk < 32; ++kk) {
            int k0 = kk * 4 + lh * 2;
            v2f a, b;
            a.x = s_h[(mt * 16 + lr) * HID + k0 + 0];
            a.y = s_h[(mt * 16 + lr) * HID + k0 + 1];
            b.x = (lr < NCLS) ? Wc3[(k0 + 0) * NCLS + lr] : 0.0f;
            b.y = (lr < NCLS) ? Wc3[(k0 + 1) * NCLS + lr] : 0.0f;
            acc = wmma_f32_k4(a, b, acc);
        }
        if (lr < NCLS) {
            float bias = bc3[lr];
            for (int v = 0; v < 8; ++v) {
                int row = mt * 16 + v + 8 * lh;
                out[row * NCLS + lr] = acc[v] + bias;
            }
        }
    }
}

extern "C" void kernel_launch(void* const* d_in, const int* in_sizes, int n_in,
                              void* d_out, int out_size, void* d_ws, size_t ws_size,
                              hipStream_t stream) {
    // setup_inputs() order
    const float* h_comp   = (const float*)d_in[0];
    const float* h_port   = (const float*)d_in[1];
    const float* h_net    = (const float*)d_in[2];
    // d_in[3..6]: edge arrays — dead code in the reference, unused.
    const int*   gid_comp = (const int*)d_in[7];
    const int*   gid_port = (const int*)d_in[8];
    const int*   gid_net  = (const int*)d_in[9];
    // d_in[10..21]: conv weights/biases — dead code, unused.
    const float* Wc1 = (const float*)d_in[22];
    const float* bc1 = (const float*)d_in[23];
    const float* Wc2 = (const float*)d_in[24];
    const float* bc2 = (const float*)d_in[25];
    const float* Wc3 = (const float*)d_in[26];
    const float* bc3 = (const float*)d_in[27];

    float* ws  = (float*)d_ws;
    float* out = (float*)d_out;

    const int nc = in_sizes[0];       // NC   (h_comp is [NC,1])
    const int np = in_sizes[1] / 2;   // NP   (h_port is [NP,2])
    const int nn = in_sizes[2];       // NN   (h_net is [NN,1])

    hipLaunchKernelGGL(zero_ws_kernel, dim3(2), dim3(256), 0, stream, ws);
    hipLaunchKernelGGL(seg_accum_kernel, dim3(256), dim3(256), 0, stream,
                       h_comp, h_port, h_net, gid_comp, gid_port, gid_net,
                       nc, np, nn, ws);
    hipLaunchKernelGGL(mlp_wmma_kernel, dim3(1), dim3(256), 0, stream,
                       ws, Wc1, bc1, Wc2, bc2, Wc3, bc3, out);
}